// ChebNet_67542655697003
// MI455X (gfx1250) — compile-verified
//
#include <hip/hip_runtime.h>

typedef __attribute__((ext_vector_type(2))) float v2f;
typedef __attribute__((ext_vector_type(8))) float v8f;

#define IN_DIM 64

// ---------------------------------------------------------------------------
// Degree accumulation: deg[row] += (row != col)
// ---------------------------------------------------------------------------
__global__ void deg_kernel(const int* __restrict__ row, const int* __restrict__ col,
                           float* __restrict__ deg, int E) {
    int e = blockIdx.x * blockDim.x + threadIdx.x;
    if (e >= E) return;
    int r = row[e], c = col[e];
    if (r != c) atomicAdd(&deg[r], 1.0f);
}

// dis[i] = deg[i] > 0 ? rsqrt(deg[i]) : 0   (in place)
__global__ void rsqrt_kernel(float* __restrict__ dis, int N) {
    int i = blockIdx.x * blockDim.x + threadIdx.x;
    if (i >= N) return;
    float d = dis[i];
    dis[i] = (d > 0.0f) ? rsqrtf(d) : 0.0f;
}

// dst = -src (elementwise), used to seed Tx2 = -Tx0 before the scale-2 SpMM
__global__ void neg_copy_kernel(const float* __restrict__ src, float* __restrict__ dst, int n) {
    int i = blockIdx.x * blockDim.x + threadIdx.x;
    if (i >= n) return;
    dst[i] = -src[i];
}

// ---------------------------------------------------------------------------
// SpMM: OUT[row] += scale * w(row,col) * V[col], w = -dis[row]*dis[col] (row!=col)
// 16 threads per edge, float4 chunks, f32 global atomics (L2-native on CDNA5).
// This is the bandwidth-bound part: ~512B of gather+atomic traffic per edge.
// ---------------------------------------------------------------------------
__global__ void spmm_kernel(const int* __restrict__ row, const int* __restrict__ col,
                            const float* __restrict__ dis, const float* __restrict__ V,
                            float* __restrict__ OUT, float scale, int E) {
    int gid = blockIdx.x * blockDim.x + threadIdx.x;
    int e = gid >> 4;
    if (e >= E) return;
    int chunk = (gid & 15) << 2;              // 0..60
    int r = row[e], c = col[e];
    if (r == c) return;
    float wgt = -scale * dis[r] * dis[c];
    if (wgt == 0.0f) return;
    const float4 v = *(const float4*)(V + (size_t)c * IN_DIM + chunk);
    float* o = OUT + (size_t)r * IN_DIM + chunk;
    atomicAdd(o + 0, wgt * v.x);
    atomicAdd(o + 1, wgt * v.y);
    atomicAdd(o + 2, wgt * v.z);
    atomicAdd(o + 3, wgt * v.w);
}

// ---------------------------------------------------------------------------
// Dense Chebyshev contraction with fp32 WMMA (V_WMMA_F32_16X16X4_F32):
//   out[16x16 tile] = sum_{s=0..2} T_s_tile(16x64) @ W[s](64xOUT_DIM) + bias
// OUT_DIM is a template constant -> all B addresses are base + immediate.
// Each wave owns one column tile, loads its 48 B-fragments ONCE into
// registers, then grid-strides over row tiles. TWO independent accumulator
// chains (even/odd K-steps) break the serial WMMA->WMMA RAW dependency so
// the scheduler keeps paired A-loads in flight and overlaps the two
// accumulation streams. __launch_bounds__(256,1) lifts the VGPR cap so all
// 48 B-fragments stay register-resident (no scratch spills in the hot loop).
// ---------------------------------------------------------------------------
template <int OUT_DIM, int DO_RELU>
__global__ void __launch_bounds__(256, 1)
cheb_gemm_wmma(const float* __restrict__ T0,
               const float* __restrict__ T1,
               const float* __restrict__ T2,
               const float* __restrict__ W,     // [3, 64, OUT_DIM]
               const float* __restrict__ bias,  // [OUT_DIM]
               float* __restrict__ out,         // [N, OUT_DIM]
               int N) {
    constexpr int COL_TILES = OUT_DIM / 16;
    const int lane = threadIdx.x & 31;
    const int wave = (blockIdx.x * blockDim.x + threadIdx.x) >> 5;
    const int totalWaves = (gridDim.x * blockDim.x) >> 5;
    const int rowWaves = totalWaves / COL_TILES;     // launch keeps this exact
    const int ct   = wave % COL_TILES;
    const int rwid = wave / COL_TILES;
    const int rowTiles = (N + 15) >> 4;

    const int n    = lane & 15;                // B/C column within tile
    const int m    = lane & 15;                // A row within tile
    const int koff = (lane >> 4) << 1;         // 0 or 2 (K split across lane halves)
    const int col0 = ct << 4;
    const int mhalf = (lane >> 4) << 3;        // 0 or 8 (C/D row split)

    // ---- load all B fragments once (W is reused for every row tile) ----
    v2f bfrag[3][16];
#pragma unroll
    for (int s = 0; s < 3; ++s) {
        const float* Ws = W + s * IN_DIM * OUT_DIM + col0 + n;
#pragma unroll
        for (int k4 = 0; k4 < 16; ++k4) {
            const int kk = k4 * 4 + koff;
            bfrag[s][k4].x = Ws[(kk + 0) * OUT_DIM];   // immediate offsets
            bfrag[s][k4].y = Ws[(kk + 1) * OUT_DIM];
        }
    }
    const float bv = bias[col0 + n];

    // ---- grid-stride over 16-row tiles; uniform loop -> EXEC all-ones ----
    for (int rt = rwid; rt < rowTiles; rt += rowWaves) {
        const int row0 = rt << 4;
        const int arow = min(row0 + m, N - 1);
        const size_t abase = (size_t)arow * IN_DIM + koff;

        v8f acc0 = {};
        v8f acc1 = {};
#pragma unroll
        for (int s = 0; s < 3; ++s) {
            const float* T = (s == 0) ? T0 : (s == 1) ? T1 : T2;
            const float* ap = T + abase;

            // Two independent chains over even/odd K-steps.
#pragma unroll
            for (int k8 = 0; k8 < 8; ++k8) {
                v2f a0, a1;
                a0.x = ap[(2 * k8 + 0) * 4 + 0];
                a0.y = ap[(2 * k8 + 0) * 4 + 1];
                a1.x = ap[(2 * k8 + 1) * 4 + 0];
                a1.y = ap[(2 * k8 + 1) * 4 + 1];
                acc0 = __builtin_amdgcn_wmma_f32_16x16x4_f32(
                    false, a0, false, bfrag[s][2 * k8 + 0], (short)0, acc0, false, false);
                acc1 = __builtin_amdgcn_wmma_f32_16x16x4_f32(
                    false, a1, false, bfrag[s][2 * k8 + 1], (short)0, acc1, false, false);
            }
        }

        // Epilogue: C/D layout — VGPR r holds row r (lanes 0-15) / r+8 (16-31)
#pragma unroll
        for (int r = 0; r < 8; ++r) {
            int mr = row0 + r + mhalf;
            if (mr < N) {
                float v = acc0[r] + acc1[r] + bv;
                if (DO_RELU) v = fmaxf(v, 0.0f);
                out[(size_t)mr * OUT_DIM + col0 + n] = v;
            }
        }
    }
}

// ---------------------------------------------------------------------------
extern "C" void kernel_launch(void* const* d_in, const int* in_sizes, int n_in,
                              void* d_out, int out_size, void* d_ws, size_t ws_size,
                              hipStream_t stream) {
    const float* x  = (const float*)d_in[0];
    const int*   ei = (const int*)d_in[1];
    const float* W1 = (const float*)d_in[2];
    const float* b1 = (const float*)d_in[3];
    const float* W2 = (const float*)d_in[4];
    const float* b2 = (const float*)d_in[5];
    float* out = (float*)d_out;

    const int N = in_sizes[0] / IN_DIM;
    const int E = in_sizes[1] / 2;
    const int* row = ei;
    const int* col = ei + E;

    float* ws  = (float*)d_ws;
    float* dis = ws;                               // N floats
    float* Tx1 = ws + N;                           // N*64
    float* Tx2 = Tx1 + (size_t)N * IN_DIM;         // N*64
    float* h   = Tx2 + (size_t)N * IN_DIM;         // N*64

    const int threads = 256;                       // 8 waves / block
    const int featN   = N * IN_DIM;
    const unsigned gridE    = (unsigned)((E + threads - 1) / threads);
    const unsigned gridN    = (unsigned)((N + threads - 1) / threads);
    const unsigned gridFeat = (unsigned)((featN + threads - 1) / threads);
    const unsigned gridSpmm = (unsigned)(((size_t)E * 16 + threads - 1) / threads);

    // ---- graph normalization ----
    hipMemsetAsync(dis, 0, (size_t)N * sizeof(float), stream);
    deg_kernel<<<gridE, threads, 0, stream>>>(row, col, dis, E);
    rsqrt_kernel<<<gridN, threads, 0, stream>>>(dis, N);

    // ---- layer 1: Tx1 = L̂ x ; Tx2 = 2 L̂ Tx1 - x ; h = relu(Σ Txk W1k + b1) ----
    hipMemsetAsync(Tx1, 0, (size_t)featN * sizeof(float), stream);
    spmm_kernel<<<gridSpmm, threads, 0, stream>>>(row, col, dis, x, Tx1, 1.0f, E);
    neg_copy_kernel<<<gridFeat, threads, 0, stream>>>(x, Tx2, featN);
    spmm_kernel<<<gridSpmm, threads, 0, stream>>>(row, col, dis, Tx1, Tx2, 2.0f, E);

    // 512 blocks * 8 waves = 4096 waves; /4 col tiles = 1024 row-waves
    cheb_gemm_wmma<64, 1><<<512, threads, 0, stream>>>(x, Tx1, Tx2, W1, b1, h, N);

    // ---- layer 2 on h, reuse Tx1/Tx2 ----
    hipMemsetAsync(Tx1, 0, (size_t)featN * sizeof(float), stream);
    spmm_kernel<<<gridSpmm, threads, 0, stream>>>(row, col, dis, h, Tx1, 1.0f, E);
    neg_copy_kernel<<<gridFeat, threads, 0, stream>>>(h, Tx2, featN);
    spmm_kernel<<<gridSpmm, threads, 0, stream>>>(row, col, dis, Tx1, Tx2, 2.0f, E);

    // 128 blocks * 8 waves = 1024 waves; 1 col tile
    cheb_gemm_wmma<16, 0><<<128, threads, 0, stream>>>(h, Tx1, Tx2, W2, b2, out, N);
}